// BahdanauAttention_40175124087272
// MI455X (gfx1250) — compile-verified
//
#include <hip/hip_runtime.h>

#define B_   4
#define TQ_  512
#define TV_  1024
#define DQ_  256
#define DV_  256
#define U_   64

typedef __attribute__((ext_vector_type(2))) float v2f;
typedef __attribute__((ext_vector_type(8))) float v8f;

// Native gfx1250 tanh (TRANS pipe). Trailing v_nop covers the multicycle
// TRANS result hazard (ISA 7.4.3) since the compiler can't schedule into asm.
__device__ __forceinline__ float fast_tanh(float x) {
  float r;
  asm volatile("v_tanh_f32 %0, %1\n\tv_nop" : "=v"(r) : "v"(x));
  return r;
}

// ---------------------------------------------------------------------------
// Kernel 1: projections.  rows [0, B*TQ) -> qproj = query@w1
//                         rows [B*TQ, B*TQ+B*TV) -> kproj = value@w2
// One 64-thread block (2 waves) per row; input row staged in LDS (broadcast),
// weight reads are lane-contiguous (coalesced 128B per wave).
// ---------------------------------------------------------------------------
__global__ __launch_bounds__(64) void proj_kernel(
    const float* __restrict__ query, const float* __restrict__ value,
    const float* __restrict__ w1,    const float* __restrict__ w2,
    float* __restrict__ qproj, float* __restrict__ kproj) {
  __shared__ float row[DQ_];
  int r = blockIdx.x;
  const float* in; const float* W; float* out;
  if (r < B_ * TQ_) {
    in = query + (size_t)r * DQ_;  W = w1;  out = qproj + (size_t)r * U_;
  } else {
    int rr = r - B_ * TQ_;
    in = value + (size_t)rr * DV_; W = w2;  out = kproj + (size_t)rr * U_;
  }
  int t = threadIdx.x;
  for (int d = t; d < DQ_; d += 64) row[d] = in[d];
  __syncthreads();
  float acc = 0.f;
#pragma unroll 8
  for (int d = 0; d < DQ_; ++d) acc = fmaf(row[d], W[d * U_ + t], acc);
  out[t] = acc;
}

// ---------------------------------------------------------------------------
// Kernel 2: scores + softmax, fused per (b, q) row.
// 256 threads (8 waves); each thread owns 4 of the 1024 v positions.
// Dominant loop: 64 native v_tanh_f32 + FMAs per v.  q-row and scale are
// broadcast from LDS; k rows read as float4 (global_load_b128).
// ---------------------------------------------------------------------------
__global__ __launch_bounds__(256) void score_softmax_kernel(
    const float* __restrict__ qproj, const float* __restrict__ kproj,
    const float* __restrict__ scale, float* __restrict__ attn) {
  __shared__ float qrow[U_];
  __shared__ float sc[U_];
  __shared__ float sv[TV_];
  __shared__ float red[256];

  int row = blockIdx.x;          // b*TQ + qi
  int b   = row / TQ_;
  int t   = threadIdx.x;
  if (t < U_) { qrow[t] = qproj[(size_t)row * U_ + t]; sc[t] = scale[t]; }
  __syncthreads();

  const float* kb = kproj + (size_t)b * TV_ * U_;
  float lmax = -3.402823466e38f;
#pragma unroll
  for (int i = 0; i < 4; ++i) {
    int v = t + i * 256;
    const float4* kr = (const float4*)(kb + (size_t)v * U_);
    const float4* q4 = (const float4*)qrow;
    const float4* s4 = (const float4*)sc;
    float s = 0.f;
#pragma unroll
    for (int u = 0; u < U_ / 4; ++u) {
      float4 k = kr[u]; float4 q = q4[u]; float4 w = s4[u];
      s = fmaf(w.x, fast_tanh(q.x + k.x), s);
      s = fmaf(w.y, fast_tanh(q.y + k.y), s);
      s = fmaf(w.z, fast_tanh(q.z + k.z), s);
      s = fmaf(w.w, fast_tanh(q.w + k.w), s);
    }
    sv[v] = s;
    lmax  = fmaxf(lmax, s);
  }

  red[t] = lmax; __syncthreads();
  for (int s = 128; s > 0; s >>= 1) {
    if (t < s) red[t] = fmaxf(red[t], red[t + s]);
    __syncthreads();
  }
  float m = red[0]; __syncthreads();

  float lsum = 0.f;
#pragma unroll
  for (int i = 0; i < 4; ++i) {
    int v = t + i * 256;
    float e = __builtin_exp2f((sv[v] - m) * 1.4426950408889634f); // v_exp_f32
    sv[v] = e; lsum += e;
  }
  red[t] = lsum; __syncthreads();
  for (int s = 128; s > 0; s >>= 1) {
    if (t < s) red[t] += red[t + s];
    __syncthreads();
  }
  float inv = 1.0f / red[0];

  float* arow = attn + (size_t)row * TV_;
#pragma unroll
  for (int i = 0; i < 4; ++i) { int v = t + i * 256; arow[v] = sv[v] * inv; }
}

// ---------------------------------------------------------------------------
// Kernel 3: context[b] = attn[b] (512x1024) @ value[b] (1024x256) via
// V_WMMA_F32_16X16X4_F32.  One wave per 16x16 C tile; 4 waves/block span
// 64 columns.  Branch-free lane addressing keeps EXEC all-ones (WMMA req).
//   A 16x4 layout: lanes 0-15 K={0,1}, lanes 16-31 K={2,3}   (ISA 7.12.2)
//   B 4x16 layout: mirrored; C/D: VGPR j = row j (lanes<16) / row 8+j.
// ---------------------------------------------------------------------------
__global__ __launch_bounds__(128) void context_kernel(
    const float* __restrict__ attn, const float* __restrict__ value,
    float* __restrict__ ctx) {
  int b    = blockIdx.z;
  int m0   = blockIdx.y * 16;
  int wave = threadIdx.x >> 5;
  int lane = threadIdx.x & 31;
  int n0   = (blockIdx.x * 4 + wave) * 16;
  int half = lane >> 4;     // 0: lanes 0-15, 1: lanes 16-31
  int lm   = lane & 15;

  const float* A = attn  + (size_t)b * TQ_ * TV_;   // [TQ, TV] row-major
  const float* V = value + (size_t)b * TV_ * DV_;   // [TV, DV] row-major

  const float* arow = A + (size_t)(m0 + lm) * TV_ + 2 * half;     // A[m][k+2h..]
  const float* vcol = V + (size_t)(2 * half) * DV_ + (n0 + lm);   // V[k+2h][n]

  v8f c = {};
  for (int k = 0; k < TV_; k += 4) {
    v2f a, bb;
    a.x  = arow[k + 0];                 // merged into global_load_b64
    a.y  = arow[k + 1];
    bb.x = vcol[(size_t)k * DV_];
    bb.y = vcol[(size_t)(k + 1) * DV_];
    c = __builtin_amdgcn_wmma_f32_16x16x4_f32(
        /*neg_a=*/false, a, /*neg_b=*/false, bb,
        /*c_mod=*/(short)0, c, /*reuse_a=*/false, /*reuse_b=*/false);
  }

  float* crow = ctx + (size_t)b * TQ_ * DV_ + (size_t)(m0 + 8 * half) * DV_ + (n0 + lm);
#pragma unroll
  for (int j = 0; j < 8; ++j) crow[(size_t)j * DV_] = c[j];
}

// ---------------------------------------------------------------------------
extern "C" void kernel_launch(void* const* d_in, const int* in_sizes, int n_in,
                              void* d_out, int out_size, void* d_ws, size_t ws_size,
                              hipStream_t stream) {
  (void)in_sizes; (void)n_in; (void)out_size; (void)ws_size;
  const float* query = (const float*)d_in[0];
  const float* value = (const float*)d_in[1];
  const float* w1    = (const float*)d_in[2];
  const float* w2    = (const float*)d_in[3];
  const float* scale = (const float*)d_in[4];

  float* out  = (float*)d_out;
  float* ctx  = out;                              // [B,TQ,DV]
  float* attn = out + (size_t)B_ * TQ_ * DV_;     // [B,TQ,TV]

  float* qproj = (float*)d_ws;                          // B*TQ*U  = 512 KB
  float* kproj = qproj + (size_t)B_ * TQ_ * U_;         // B*TV*U  = 1 MB

  proj_kernel<<<B_ * TQ_ + B_ * TV_, 64, 0, stream>>>(
      query, value, w1, w2, qproj, kproj);

  score_softmax_kernel<<<B_ * TQ_, 256, 0, stream>>>(
      qproj, kproj, scale, attn);

  context_kernel<<<dim3(DV_ / 64, TQ_ / 16, B_), 128, 0, stream>>>(
      attn, value, ctx);
}